// CinST_weight_59304908423199
// MI455X (gfx1250) — compile-verified
//
#include <hip/hip_runtime.h>
#include <hip/hip_bf16.h>
#include <float.h>

// ---------------------------------------------------------------------------
// Shapes: x (64,256,56,56) fp32, T=8 -> b=8.  HW=3136, THW=25088, CHW=802816.
// Memory-bound problem (~615 MB total traffic -> ~30us @ 23.3 TB/s HBM).
// All intermediates (~101 MB) live in d_ws and fit in the 192 MB L2.
// ---------------------------------------------------------------------------

typedef __attribute__((ext_vector_type(2))) float v2f;
typedef __attribute__((ext_vector_type(8))) float v8f;
typedef int vs4i __attribute__((vector_size(16)));   // matches builtin param

#define HW   3136
#define THW  25088
#define NC   256
#define NT   8
#define NB   8

// gfx1250 async memory->LDS path (ASYNCcnt-tracked), guarded so the TU still
// compiles (with the plain load+ds_store fallback) if the toolchain lacks it.
#if defined(__has_builtin)
#if __has_builtin(__builtin_amdgcn_global_load_async_to_lds_b128) && \
    __has_builtin(__builtin_amdgcn_s_wait_asynccnt)
#define USE_ASYNC_LDS 1
#endif
#endif

// ---------------------------------------------------------------------------
// Kernel 1: channel pool.  pc[b][{max,mean}][t][h][w] over C=256.
// mean computed with V_WMMA_F32_16X16X4_F32: D = ones(16x4) x B(4x16) + C,
// accumulated over 64 chunks of 4 channels.  Sum over all k-chunks is
// invariant to the hardware's k-slot ordering since A is all ones; only
// N = lane%16 striping (documented for B/C/D) matters.
// One wave handles 16 pixels.  Block = 128 threads = 4 waves = 64 pixels.
// EXEC is all-ones through the whole WMMA loop (no divergence before it).
// ---------------------------------------------------------------------------
__global__ void pool_c_wmma(const float* __restrict__ x, float* __restrict__ pc)
{
    const int bn   = blockIdx.x;          // 0..63  (= b*8 + t)
    const int b    = bn >> 3;
    const int t    = bn & 7;
    const int lane = threadIdx.x & 31;
    const int wave = threadIdx.x >> 5;
    const int n     = lane & 15;          // pixel within group of 16
    const int khalf = lane >> 4;          // 0 -> k{0,1}, 1 -> k{2,3}
    const int pix   = blockIdx.y * 64 + wave * 16 + n;   // 0..3135 exactly

    const float* p0 = x + bn * (NC * HW) + (2 * khalf) * HW + pix;

    v2f a = {1.0f, 1.0f};                 // ones A-matrix (16x4)
    v8f acc = {};
    float mx = -FLT_MAX;

    for (int c0 = 0; c0 < NC; c0 += 4) {
        float b0 = p0[c0 * HW];           // channel c0 + 2*khalf
        float b1 = p0[c0 * HW + HW];      // channel c0 + 2*khalf + 1
        mx = fmaxf(mx, fmaxf(b0, b1));
        v2f bv = {b0, b1};
        acc = __builtin_amdgcn_wmma_f32_16x16x4_f32(
                  false, a, false, bv, (short)0, acc, false, false);
    }
    // D row M=0 (lanes 0-15) / M=8 (lanes 16-31) both hold the full column sum.
    float mean = acc[0] * (1.0f / 256.0f);
    // lanes L and L+16 covered disjoint channel halves of the same pixel:
    mx = fmaxf(mx, __shfl_xor(mx, 16, 32));

    if (lane < 16) {
        // pc layout (b, 2, T, H, W); pool index 0 = max, 1 = mean
        pc[((b * 2 + 0) * NT + t) * HW + pix] = mx;
        pc[((b * 2 + 1) * NT + t) * HW + pix] = mean;
    }
}

// ---------------------------------------------------------------------------
// Generic 2->1 channel 3x3x3 SAME conv (cross-correlation, like lax.conv /
// torch) over (D,H,W) of an input shaped (8, 2, D, H, W), followed by
// eval-mode BN (mean=0,var=1) + sigmoid.  Used for all four gate convs.
// ---------------------------------------------------------------------------
__global__ void conv3_bn_sig(const float* __restrict__ in,
                             const float* __restrict__ wgt,   // 54 floats
                             const float* __restrict__ g4,
                             const float* __restrict__ b4,
                             int gi,
                             float* __restrict__ out,
                             int D, int H, int W)
{
    const int total = NB * D * H * W;
    const int idx = blockIdx.x * blockDim.x + threadIdx.x;
    if (idx >= total) return;

    int w = idx % W;
    int tmp = idx / W;
    int h = tmp % H;  tmp /= H;
    int d = tmp % D;
    int b = tmp / D;

    float wr[54];
#pragma unroll
    for (int i = 0; i < 54; ++i) wr[i] = wgt[i];   // uniform -> scalar loads

    const int HWp = H * W;
    const int DHW = D * HWp;
    const float* base = in + b * 2 * DHW;

    float s = 0.0f;
    if (d > 0 && d < D - 1 && h > 0 && h < H - 1 && w > 0 && w < W - 1) {
#pragma unroll
        for (int i = 0; i < 2; ++i)
#pragma unroll
            for (int kd = 0; kd < 3; ++kd)
#pragma unroll
                for (int kh = 0; kh < 3; ++kh)
#pragma unroll
                    for (int kw = 0; kw < 3; ++kw)
                        s += wr[((i * 3 + kd) * 3 + kh) * 3 + kw] *
                             base[i * DHW + (d + kd - 1) * HWp +
                                  (h + kh - 1) * W + (w + kw - 1)];
    } else {
        for (int i = 0; i < 2; ++i)
            for (int kd = 0; kd < 3; ++kd) {
                int dd = d + kd - 1; if (dd < 0 || dd >= D) continue;
                for (int kh = 0; kh < 3; ++kh) {
                    int hh = h + kh - 1; if (hh < 0 || hh >= H) continue;
                    for (int kw = 0; kw < 3; ++kw) {
                        int ww = w + kw - 1; if (ww < 0 || ww >= W) continue;
                        s += wr[((i * 3 + kd) * 3 + kh) * 3 + kw] *
                             base[i * DHW + dd * HWp + hh * W + ww];
                    }
                }
            }
    }
    const float scale = g4[gi] / sqrtf(1.0f + 1e-5f);
    const float z = s * scale + b4[gi];
    out[idx] = 1.0f / (1.0f + __expf(-z));
}

// ---------------------------------------------------------------------------
// Kernel 3: fused gate-apply + T/H/W pools.  One block per (b,c); stages the
// full gated (T,H,W) slice (100,352 B) in LDS (dynamic, well under the 320 KB
// per-WGP budget), then all three pools read LDS only -> x is read exactly
// once here.  The stage uses gfx1250 GLOBAL_LOAD_ASYNC_TO_LDS_B128 (ASYNCcnt)
// when available: memory -> LDS without bouncing through VGPRs.
// ---------------------------------------------------------------------------
__global__ void fused_pools(const float* __restrict__ x,
                            const float* __restrict__ yc,
                            float* __restrict__ pt,    // (b,2,C,H,W)
                            float* __restrict__ ph,    // (b,2,T,C,W)
                            float* __restrict__ pw)    // (b,2,T,H,C)
{
    extern __shared__ float xs[];          // 25088 floats
    const int c = blockIdx.x;              // 0..255
    const int b = blockIdx.y;              // 0..7
    const int tid = threadIdx.x;           // 0..255

#if defined(USE_ASYNC_LDS)
    // async DMA the raw x slice into LDS, 16B per op (6272 ops per block)
    for (int i = tid; i < THW / 4; i += 256) {
        int t = i / (HW / 4);
        int r = i - t * (HW / 4);
        const float* g = x + ((b * NT + t) * NC + c) * HW + r * 4;
        float* l = xs + t * HW + r * 4;
        __builtin_amdgcn_global_load_async_to_lds_b128(
            (__attribute__((address_space(1))) vs4i*)g,
            (__attribute__((address_space(3))) vs4i*)l, 0, 0);
    }
    __builtin_amdgcn_s_wait_asynccnt(0);   // this wave's LDS writes landed
    __syncthreads();                       // all waves' writes landed
    // apply the channel gate in place: xc = x * yc
    for (int idx = tid; idx < THW; idx += 256)
        xs[idx] *= yc[b * THW + idx];
#else
    for (int idx = tid; idx < THW; idx += 256) {
        int t  = idx / HW;
        int hw = idx - t * HW;
        xs[idx] = x[((b * NT + t) * NC + c) * HW + hw] * yc[b * THW + idx];
    }
#endif
    __syncthreads();

    // pool over T -> pt[b][p][c][hw]
    for (int pix = tid; pix < HW; pix += 256) {
        float mx = -FLT_MAX, sm = 0.0f;
#pragma unroll
        for (int t = 0; t < NT; ++t) {
            float v = xs[t * HW + pix];
            mx = fmaxf(mx, v);  sm += v;
        }
        pt[((b * 2 + 0) * NC + c) * HW + pix] = mx;
        pt[((b * 2 + 1) * NC + c) * HW + pix] = sm * 0.125f;
    }
    // pool over H -> ph[b][p][t][c][w]
    for (int i = tid; i < NT * 56; i += 256) {
        int t = i / 56, w = i - t * 56;
        float mx = -FLT_MAX, sm = 0.0f;
        for (int h = 0; h < 56; ++h) {
            float v = xs[t * HW + h * 56 + w];
            mx = fmaxf(mx, v);  sm += v;
        }
        ph[(((b * 2 + 0) * NT + t) * NC + c) * 56 + w] = mx;
        ph[(((b * 2 + 1) * NT + t) * NC + c) * 56 + w] = sm * (1.0f / 56.0f);
    }
    // pool over W -> pw[b][p][t][h][c]
    for (int i = tid; i < NT * 56; i += 256) {
        int t = i / 56, h = i - t * 56;
        float mx = -FLT_MAX, sm = 0.0f;
        for (int w = 0; w < 56; ++w) {
            float v = xs[t * HW + h * 56 + w];
            mx = fmaxf(mx, v);  sm += v;
        }
        pw[(((b * 2 + 0) * NT + t) * 56 + h) * NC + c] = mx;
        pw[(((b * 2 + 1) * NT + t) * 56 + h) * NC + c] = sm * (1.0f / 56.0f);
    }
}

// ---------------------------------------------------------------------------
// Kernel 7: out = (yt + yh + yw) / 3, broadcast to (64,256,56,56).
// float4 over W (56 % 4 == 0); this is the single 205 MB store.
// ---------------------------------------------------------------------------
__global__ void combine_gates(const float* __restrict__ yt,   // (b,C,H,W)
                              const float* __restrict__ yh,   // (b,T,C,W)
                              const float* __restrict__ yw,   // (b,T,H,C)
                              float* __restrict__ out)
{
    const int n4 = 64 * NC * HW / 4;                 // 12,845,056
    int i4 = blockIdx.x * 256 + threadIdx.x;
    if (i4 >= n4) return;
    int f    = i4 * 4;
    int hw   = f % HW;
    int rest = f / HW;
    int c  = rest % NC;
    int bt = rest / NC;
    int t  = bt & 7;
    int b  = bt >> 3;
    int h  = hw / 56;
    int w  = hw % 56;

    // yw gather is the only non-streaming read here; prefetch its line
    // (lowers to global_prefetch_b8 on gfx1250).
    __builtin_prefetch(yw + ((b * NT + t) * 56 + h) * NC + c, 0, 1);

    const float4 a  = *(const float4*)(yt + (b * NC + c) * HW + hw);
    const float4 bh = *(const float4*)(yh + ((b * NT + t) * NC + c) * 56 + w);
    const float  cw = yw[((b * NT + t) * 56 + h) * NC + c];

    float4 o;
    o.x = (a.x + bh.x + cw) * (1.0f / 3.0f);
    o.y = (a.y + bh.y + cw) * (1.0f / 3.0f);
    o.z = (a.z + bh.z + cw) * (1.0f / 3.0f);
    o.w = (a.w + bh.w + cw) * (1.0f / 3.0f);
    *(float4*)(out + f) = o;
}

// ---------------------------------------------------------------------------
extern "C" void kernel_launch(void* const* d_in, const int* in_sizes, int n_in,
                              void* d_out, int out_size, void* d_ws, size_t ws_size,
                              hipStream_t stream)
{
    const float* x     = (const float*)d_in[0];
    const float* Wc    = (const float*)d_in[1];
    const float* Wt    = (const float*)d_in[2];
    const float* Wh    = (const float*)d_in[3];
    const float* Ww    = (const float*)d_in[4];
    const float* gamma = (const float*)d_in[5];
    const float* beta  = (const float*)d_in[6];
    float* out = (float*)d_out;

    // workspace carve-up (floats); total 25,374,720 floats = ~101.5 MB
    float* ws = (float*)d_ws;
    float* pc = ws;                 // (8,2,8,56,56)   =   401,408
    float* yc = pc + 401408;        // (8,8,56,56)     =   200,704
    float* pt = yc + 200704;        // (8,2,256,56,56) = 12,845,056
    float* ph = pt + 12845056;      // (8,2,8,256,56)  =  1,835,008
    float* pw = ph + 1835008;       // (8,2,8,56,256)  =  1,835,008
    float* yt = pw + 1835008;       // (8,256,56,56)   =  6,422,528
    float* yh = yt + 6422528;       // (8,8,256,56)    =    917,504
    float* yw = yh + 917504;        // (8,8,56,256)    =    917,504

    // 1) channel pool via WMMA f32 16x16x4
    pool_c_wmma<<<dim3(64, 49), 128, 0, stream>>>(x, pc);

    // 2) channel-gate conv -> yc (b,T,H,W)
    conv3_bn_sig<<<(NB * NT * HW + 255) / 256, 256, 0, stream>>>(
        pc, Wc, gamma, beta, 0, yc, NT, 56, 56);

    // 3) gate + fused T/H/W pools (100,352 B dynamic LDS per block)
    fused_pools<<<dim3(NC, NB), 256, THW * sizeof(float), stream>>>(
        x, yc, pt, ph, pw);

    // 4) temporal gate conv -> yt (b,C,H,W)
    conv3_bn_sig<<<(NB * NC * HW + 255) / 256, 256, 0, stream>>>(
        pt, Wt, gamma, beta, 1, yt, NC, 56, 56);

    // 5) height gate conv -> yh (b,T,C,W)
    conv3_bn_sig<<<(NB * NT * NC * 56 + 255) / 256, 256, 0, stream>>>(
        ph, Wh, gamma, beta, 2, yh, NT, NC, 56);

    // 6) width gate conv -> yw (b,T,H,C)
    conv3_bn_sig<<<(NB * NT * 56 * NC + 255) / 256, 256, 0, stream>>>(
        pw, Ww, gamma, beta, 3, yw, NT, 56, NC);

    // 7) broadcast combine -> out
    combine_gates<<<(64 * NC * HW / 4 + 255) / 256, 256, 0, stream>>>(
        yt, yh, yw, out);
}